// HHGN_69793218560324
// MI455X (gfx1250) — compile-verified
//
#include <hip/hip_runtime.h>

// ---------------------------------------------------------------------------
// HHGN: 2-layer heterogeneous GraphConv, D=128, fp32.
// Scatter phase: HBM/atomic bound (~5 GB traffic) -> global_atomic_add_f32.
// GEMM phase: v_wmma_f32_16x16x4_f32 from LDS-staged tiles (exact fp32).
// ---------------------------------------------------------------------------

#define D 128

typedef __attribute__((ext_vector_type(2))) float v2f;
typedef __attribute__((ext_vector_type(8))) float v8f;

// ---- degree histograms ----------------------------------------------------
__global__ void deg_accum_kernel(const int* __restrict__ src,
                                 const int* __restrict__ dst,
                                 float* __restrict__ dout,
                                 float* __restrict__ din, int ne) {
    int e = blockIdx.x * blockDim.x + threadIdx.x;
    if (e < ne) {
        atomicAdd(&dout[src[e]], 1.0f);
        atomicAdd(&din[dst[e]], 1.0f);
    }
}

__global__ void deg_finalize_kernel(float* __restrict__ d, int n) {
    int i = blockIdx.x * blockDim.x + threadIdx.x;
    if (i < n) d[i] = rsqrtf(fmaxf(d[i], 1.0f));
}

// ---- edge scatter: one wave32 per edge, float4 per lane -------------------
__global__ void scatter_kernel(const float* __restrict__ x,
                               const float* __restrict__ rsq_out,
                               const int* __restrict__ src,
                               const int* __restrict__ dst,
                               float* __restrict__ agg, int ne) {
    int wid  = (blockIdx.x * blockDim.x + threadIdx.x) >> 5;
    int lane = threadIdx.x & 31;
    if (wid >= ne) return;
    int s = src[wid], d = dst[wid];
    float sc = rsq_out[s];
    const float4 v = ((const float4*)(x + (size_t)s * D))[lane];
    float* base = agg + (size_t)d * D + lane * 4;
    atomicAdd(base + 0, v.x * sc);
    atomicAdd(base + 1, v.y * sc);
    atomicAdd(base + 2, v.z * sc);
    atomicAdd(base + 3, v.w * sc);
}

// ---- WMMA GEMM: out[tile 16 x 64] = (agg * rsq_in) @ W + b ---------------
// Block: 128 threads = 4 waves; wave w owns 16 output columns.
// LDS: W slice 128x64 (32 KB) + scaled A tile 16x128 (8 KB).
__global__ void __launch_bounds__(128)
gemm_tile_kernel(const float* __restrict__ agg,
                 const float* __restrict__ rsq_in,
                 const float* __restrict__ W,
                 const float* __restrict__ bias,
                 float* __restrict__ out, int n, int accumulate) {
    __shared__ __align__(16) float sW[D * 64];
    __shared__ __align__(16) float sA[16 * D];

    const int tid   = threadIdx.x;
    const int lane  = tid & 31;
    const int wv    = tid >> 5;            // 0..3
    const int m0    = blockIdx.x * 16;
    const int nbase = blockIdx.y * 64;

    // Stage W[:, nbase:nbase+64]  (2048 float4, 16 per thread)
    for (int i = tid; i < 2048; i += 128) {
        int r = i >> 4;                    // 16 float4 per 64-col row
        int c = (i & 15) << 2;
        *(float4*)&sW[r * 64 + c] =
            *(const float4*)&W[(size_t)r * D + nbase + c];
    }
    // Stage scaled A tile (16 x 128), zero-fill OOB rows (keeps EXEC clean
    // for the WMMA loop).
    for (int i = tid; i < 512; i += 128) {
        int r = i >> 5;
        int c = (i & 31) << 2;
        int row = m0 + r;
        float4 v = {0.f, 0.f, 0.f, 0.f};
        if (row < n) {
            float s = rsq_in[row];
            v = *(const float4*)&agg[(size_t)row * D + c];
            v.x *= s; v.y *= s; v.z *= s; v.w *= s;
        }
        *(float4*)&sA[r * D + c] = v;
    }
    __syncthreads();

    const int n0 = wv * 16;
    const int nc = lane & 15;              // N (cols) / A-row index
    const int hi = lane >> 4;              // lane group: K pair select

    v8f acc = {};
#pragma unroll
    for (int k = 0; k < D; k += 4) {
        // A 16x4 layout: lanes 0-15 -> K={k,k+1}, lanes 16-31 -> K={k+2,k+3}
        v2f a, b;
        a.x = sA[nc * D + k + 2 * hi + 0];
        a.y = sA[nc * D + k + 2 * hi + 1];
        // B 4x16 layout: VGPR0 -> K=k (+2 for hi lanes), VGPR1 -> K=k+1 (+2)
        b.x = sW[(k + 2 * hi + 0) * 64 + n0 + nc];
        b.y = sW[(k + 2 * hi + 1) * 64 + n0 + nc];
        acc = __builtin_amdgcn_wmma_f32_16x16x4_f32(
            false, a, false, b, (short)0, acc, false, false);
    }

    // C/D layout: VGPR r -> M=r (lanes 0-15) / M=r+8 (lanes 16-31)
    float bv = bias[nbase + n0 + nc];
#pragma unroll
    for (int r = 0; r < 8; ++r) {
        int row = m0 + r + 8 * hi;
        if (row < n) {
            size_t idx = (size_t)row * D + nbase + n0 + nc;
            float val = acc[r] + bv;
            if (accumulate) val += out[idx];
            out[idx] = val;
        }
    }
}

// ---------------------------------------------------------------------------
extern "C" void kernel_launch(void* const* d_in, const int* in_sizes, int n_in,
                              void* d_out, int out_size, void* d_ws, size_t ws_size,
                              hipStream_t stream) {
    (void)n_in; (void)out_size; (void)ws_size;

    const float* h_a    = (const float*)d_in[0];
    const float* h_t    = (const float*)d_in[1];
    const int*   aa_src = (const int*)d_in[2];
    const int*   aa_dst = (const int*)d_in[3];
    const int*   at_src = (const int*)d_in[4];
    const int*   at_dst = (const int*)d_in[5];
    const int*   ta_src = (const int*)d_in[6];
    const int*   ta_dst = (const int*)d_in[7];
    const float* W_aa   = (const float*)d_in[8];
    const float* b_aa   = (const float*)d_in[9];
    const float* W_at   = (const float*)d_in[10];
    const float* b_at   = (const float*)d_in[11];
    const float* W_ta   = (const float*)d_in[12];
    const float* b_ta   = (const float*)d_in[13];

    const int NA  = in_sizes[0] / D;
    const int NT  = in_sizes[1] / D;
    const int Eaa = in_sizes[2];
    const int Eat = in_sizes[4];
    const int Eta = in_sizes[6];

    // ---- workspace layout (floats) ----
    float* ws = (float*)d_ws;
    size_t off = 0;
    float* rsq_out_aa = ws + off; off += NA;
    float* rsq_in_aa  = ws + off; off += NA;
    float* rsq_out_at = ws + off; off += NA;
    float* rsq_in_at  = ws + off; off += NT;
    float* rsq_out_ta = ws + off; off += NT;
    float* rsq_in_ta  = ws + off; off += NA;
    const size_t deg_total = off;
    float* h_a1 = ws + off; off += (size_t)NA * D;
    float* h_t1 = ws + off; off += (size_t)NT * D;
    float* aggA = ws + off; off += (size_t)NA * D;
    float* aggT = ws + off; off += (size_t)NT * D;

    // ---- degrees (edge-structure only; recomputed deterministically) ----
    hipMemsetAsync(ws, 0, deg_total * sizeof(float), stream);
    deg_accum_kernel<<<(Eaa + 255) / 256, 256, 0, stream>>>(aa_src, aa_dst, rsq_out_aa, rsq_in_aa, Eaa);
    deg_accum_kernel<<<(Eat + 255) / 256, 256, 0, stream>>>(at_src, at_dst, rsq_out_at, rsq_in_at, Eat);
    deg_accum_kernel<<<(Eta + 255) / 256, 256, 0, stream>>>(ta_src, ta_dst, rsq_out_ta, rsq_in_ta, Eta);
    deg_finalize_kernel<<<((int)deg_total + 255) / 256, 256, 0, stream>>>(ws, (int)deg_total);

    auto relation = [&](const float* xin, const float* rsq_o, const float* rsq_i,
                        const int* src, const int* dst, int ne, int ndst,
                        const float* Wm, const float* bm, float* agg,
                        float* outp, int accflag) {
        hipMemsetAsync(agg, 0, (size_t)ndst * D * sizeof(float), stream);
        int blocks = (ne + 7) / 8;  // 8 waves per 256-thread block, 1 edge/wave
        scatter_kernel<<<blocks, 256, 0, stream>>>(xin, rsq_o, src, dst, agg, ne);
        dim3 g((ndst + 15) / 16, 2);
        gemm_tile_kernel<<<g, 128, 0, stream>>>(agg, rsq_i, Wm, bm, outp, ndst, accflag);
    };

    // ---- layer 1: (h_a, h_t) -> (h_a1, h_t1) ----
    relation(h_a, rsq_out_aa, rsq_in_aa, aa_src, aa_dst, Eaa, NA, W_aa, b_aa, aggA, h_a1, 0);
    relation(h_t, rsq_out_ta, rsq_in_ta, ta_src, ta_dst, Eta, NA, W_ta, b_ta, aggA, h_a1, 1);
    relation(h_a, rsq_out_at, rsq_in_at, at_src, at_dst, Eat, NT, W_at, b_at, aggT, h_t1, 0);

    // ---- layer 2: (h_a1, h_t1) -> d_out ----
    float* out_a = (float*)d_out;
    float* out_t = out_a + (size_t)NA * D;
    relation(h_a1, rsq_out_aa, rsq_in_aa, aa_src, aa_dst, Eaa, NA, W_aa, b_aa, aggA, out_a, 0);
    relation(h_t1, rsq_out_ta, rsq_in_ta, ta_src, ta_dst, Eta, NA, W_ta, b_ta, aggA, out_a, 1);
    relation(h_a1, rsq_out_at, rsq_in_at, at_src, at_dst, Eat, NT, W_at, b_at, aggT, out_t, 0);
}